// GraphConvAutoEncoder_VertexRecon_24601572671834
// MI455X (gfx1250) — compile-verified
//
#include <hip/hip_runtime.h>
#include <math.h>

// GCN: 3 layers of  (h @ W) -> edge gather*w -> segment_sum -> +b [-> sigmoid]
// N=50000 (= 3125*16 exactly), E=500000, feats 128 -> 64 -> 128 -> 32.
// Output tuple: out [N,32] then feat [N,128], concatenated in d_out.

typedef float v2f __attribute__((ext_vector_type(2)));
typedef float v8f __attribute__((ext_vector_type(8)));

#define GCN_NFEAT 128
#define GCN_NHID 64
#define GCN_NCLASS 32

// ---------------------------------------------------------------------------
// WMMA f32 GEMM:  C[nrows x NCOLS] = A[nrows x K] @ W[K x NCOLS]
// One wave (32 threads) per 16-row tile; computes all NCOLS/16 column tiles,
// reusing the A fragment across the column tiles each K-step.
//
// v_wmma_f32_16x16x4_f32 fragment layout (wave32, ISA 7.12.2):
//   A (16x4): lane l<16 holds A[M=l][K=kb..kb+1] with kb = (l>=16 ? 2 : 0)
//   B (4x16): lane l<16 holds B[K=kb..kb+1][N=l]   (mirror of A)
//   C/D (16x16, 8 VGPRs): vgpr r, lanes 0-15 -> M=r, N=lane;
//                                  lanes 16-31 -> M=8+r, N=lane-16
// ---------------------------------------------------------------------------
template <int K, int NCOLS>
__global__ __launch_bounds__(32) void gemm_wmma_f32(
    const float* __restrict__ A, const float* __restrict__ W,
    float* __restrict__ C) {
  constexpr int NT = NCOLS / 16;
  const int rowTile = blockIdx.x;
  const int lane = threadIdx.x;
  const int half = lane >> 4;  // 0 or 1
  const int l15 = lane & 15;
  const int kb = half * 2;

  const float* arow = A + (size_t)(rowTile * 16 + l15) * K;

  v8f acc[NT];
#pragma unroll
  for (int t = 0; t < NT; ++t) acc[t] = (v8f){0.f, 0.f, 0.f, 0.f, 0.f, 0.f, 0.f, 0.f};

  for (int k = 0; k < K; k += 4) {
    // A fragment: two consecutive fp32 along K (8-byte aligned: K % 2 == 0)
    v2f a = *(const v2f*)(arow + k + kb);
#pragma unroll
    for (int t = 0; t < NT; ++t) {
      v2f b;
      b.x = W[(size_t)(k + kb) * NCOLS + t * 16 + l15];
      b.y = W[(size_t)(k + kb + 1) * NCOLS + t * 16 + l15];
      // 8 args: (neg_a, A, neg_b, B, c_mod, C, reuse_a, reuse_b)
      acc[t] = __builtin_amdgcn_wmma_f32_16x16x4_f32(
          false, a, false, b, (short)0, acc[t], false, false);
    }
  }

#pragma unroll
  for (int t = 0; t < NT; ++t) {
    float* cbase = C + (size_t)(rowTile * 16) * NCOLS + t * 16 + l15;
#pragma unroll
    for (int r = 0; r < 8; ++r) {
      cbase[(size_t)(r + 8 * half) * NCOLS] = acc[t][r];
    }
  }
}

// ---------------------------------------------------------------------------
// Edge scatter: agg[dst[e], :] += support[src[e], :] * w[e]
// float4 per work-item; consecutive threads cover consecutive feature chunks
// of the same edge (coalesced gather). Atomic adds land in L2 (agg fits).
// ---------------------------------------------------------------------------
template <int F>
__global__ __launch_bounds__(256) void scatter_edges(
    const float* __restrict__ sup, const int* __restrict__ src,
    const int* __restrict__ dst, const float* __restrict__ w,
    float* __restrict__ agg, int E) {
  constexpr int CH = F / 4;
  const long long total = (long long)E * CH;
  for (long long i = (long long)blockIdx.x * blockDim.x + threadIdx.x;
       i < total; i += (long long)gridDim.x * blockDim.x) {
    const int e = (int)(i / CH);
    const int c = (int)(i % CH);
    const float we = w[e];
    const int s = src[e];
    const int d = dst[e];
    const float4 v = *(const float4*)(sup + (size_t)s * F + c * 4);
    float* p = agg + (size_t)d * F + c * 4;
    unsafeAtomicAdd(p + 0, v.x * we);
    unsafeAtomicAdd(p + 1, v.y * we);
    unsafeAtomicAdd(p + 2, v.z * we);
    unsafeAtomicAdd(p + 3, v.w * we);
  }
}

__global__ __launch_bounds__(256) void zero_buf(float* __restrict__ p, long long n) {
  for (long long i = (long long)blockIdx.x * blockDim.x + threadIdx.x; i < n;
       i += (long long)gridDim.x * blockDim.x)
    p[i] = 0.0f;
}

// out[i] = bias[i % F]  (pre-initialize accumulation target with bias)
template <int F>
__global__ __launch_bounds__(256) void init_bias(float* __restrict__ p,
                                                 const float* __restrict__ b,
                                                 long long n) {
  for (long long i = (long long)blockIdx.x * blockDim.x + threadIdx.x; i < n;
       i += (long long)gridDim.x * blockDim.x)
    p[i] = b[i % F];
}

// p[i] = sigmoid(p[i] + bias[i % F])   (in place)
template <int F>
__global__ __launch_bounds__(256) void bias_sigmoid(float* __restrict__ p,
                                                    const float* __restrict__ b,
                                                    long long n) {
  for (long long i = (long long)blockIdx.x * blockDim.x + threadIdx.x; i < n;
       i += (long long)gridDim.x * blockDim.x) {
    float v = p[i] + b[i % F];
    p[i] = 1.0f / (1.0f + __expf(-v));
  }
}

static inline int nblk(long long n, int bs) {
  long long b = (n + bs - 1) / bs;
  if (b > 1048576) b = 1048576;  // grid-stride handles the rest
  return (int)b;
}

extern "C" void kernel_launch(void* const* d_in, const int* in_sizes, int n_in,
                              void* d_out, int out_size, void* d_ws,
                              size_t ws_size, hipStream_t stream) {
  const float* x = (const float*)d_in[0];           // [N,128]
  const int* edge_index = (const int*)d_in[1];      // [2,E]
  const float* ew = (const float*)d_in[2];          // [E]
  const float* W1 = (const float*)d_in[3];          // [128,64]
  const float* b1 = (const float*)d_in[4];          // [64]
  const float* W2 = (const float*)d_in[5];          // [64,128]
  const float* b2 = (const float*)d_in[6];          // [128]
  const float* W3 = (const float*)d_in[7];          // [128,32]
  const float* b3 = (const float*)d_in[8];          // [32]

  const int N = in_sizes[0] / GCN_NFEAT;  // 50000
  const int E = in_sizes[2];              // 500000
  const int* src = edge_index;
  const int* dst = edge_index + E;

  float* out = (float*)d_out;                   // [N,32]
  float* feat = out + (size_t)N * GCN_NCLASS;   // [N,128]

  float* bufA = (float*)d_ws;                   // N*64 (support1 / h? / support3)
  float* bufB = bufA + (size_t)N * GCN_NHID;    // N*64 (agg1 -> h1)
  float* bufC = bufB + (size_t)N * GCN_NHID;    // N*128 (support2)

  const int rowTiles = N / 16;  // 3125, exact

  // ---- Layer 1: support1 = x @ W1 ------------------------------------
  gemm_wmma_f32<GCN_NFEAT, GCN_NHID>
      <<<dim3(rowTiles), dim3(32), 0, stream>>>(x, W1, bufA);
  zero_buf<<<nblk((long long)N * GCN_NHID, 256), 256, 0, stream>>>(
      bufB, (long long)N * GCN_NHID);
  scatter_edges<GCN_NHID>
      <<<nblk((long long)E * (GCN_NHID / 4), 256), 256, 0, stream>>>(
          bufA, src, dst, ew, bufB, E);
  bias_sigmoid<GCN_NHID><<<nblk((long long)N * GCN_NHID, 256), 256, 0, stream>>>(
      bufB, b1, (long long)N * GCN_NHID);  // h1 in bufB

  // ---- Layer 2: support2 = h1 @ W2; agg directly into feat -----------
  gemm_wmma_f32<GCN_NHID, 2 * GCN_NHID>
      <<<dim3(rowTiles), dim3(32), 0, stream>>>(bufB, W2, bufC);
  zero_buf<<<nblk((long long)N * 2 * GCN_NHID, 256), 256, 0, stream>>>(
      feat, (long long)N * 2 * GCN_NHID);
  scatter_edges<2 * GCN_NHID>
      <<<nblk((long long)E * (2 * GCN_NHID / 4), 256), 256, 0, stream>>>(
          bufC, src, dst, ew, feat, E);
  bias_sigmoid<2 * GCN_NHID>
      <<<nblk((long long)N * 2 * GCN_NHID, 256), 256, 0, stream>>>(
          feat, b2, (long long)N * 2 * GCN_NHID);  // feat done (output #2)

  // ---- Layer 3: support3 = feat @ W3; agg into out (pre-init b3) -----
  gemm_wmma_f32<2 * GCN_NHID, GCN_NCLASS>
      <<<dim3(rowTiles), dim3(32), 0, stream>>>(feat, W3, bufA);
  init_bias<GCN_NCLASS><<<nblk((long long)N * GCN_NCLASS, 256), 256, 0, stream>>>(
      out, b3, (long long)N * GCN_NCLASS);
  scatter_edges<GCN_NCLASS>
      <<<nblk((long long)E * (GCN_NCLASS / 4), 256), 256, 0, stream>>>(
          bufA, src, dst, ew, out, E);  // out done (output #1)
}